// SimpleMHA2D_35072702939569
// MI455X (gfx1250) — compile-verified
//
#include <hip/hip_runtime.h>

typedef __attribute__((ext_vector_type(2))) float v2f;
typedef __attribute__((ext_vector_type(8))) float v8f;

static constexpr int BATCH = 32;
static constexpr int S  = 1024;   // H*W
static constexpr int KV = 1024;
static constexpr int NH = 16;     // heads
static constexpr int DK = 64;
static constexpr int DV = 64;

#define WMMA_F32_4(A, B, C) \
  __builtin_amdgcn_wmma_f32_16x16x4_f32(false, (A), false, (B), (short)0, (C), false, false)

// ---------------------------------------------------------------------------
// Kernel 1: fold query into k_kernel.
//   w[kv][n] = sum_k k_kernel[kv][n*64+k] * q[n][k]    (stored WMMA-swizzled)
//   blogit[n] = sum_k k_bias[n*64+k] * q[n][k]
// Swizzle: element (kv, n) -> wswz[(kv/4)*64 + lane*2 + r],
//          lane = n + 16*((kv%4)/2), r = (kv%4)&1
// ---------------------------------------------------------------------------
__global__ void __launch_bounds__(256) prep_w_kernel(
    const float* __restrict__ k_kernel,
    const float* __restrict__ k_bias,
    const float* __restrict__ query,
    float* __restrict__ wswz,
    float* __restrict__ blogit) {
  int t  = blockIdx.x * blockDim.x + threadIdx.x;   // [0, KV*NH)
  int kv = t >> 4;
  int n  = t & 15;
  const float* col = k_kernel + (size_t)kv * (NH * DK) + n * DK;
  const float* q   = query + n * DK;
  float acc = 0.f;
#pragma unroll 8
  for (int k = 0; k < DK; ++k) acc += col[k] * q[k];
  int kb   = kv >> 2;
  int kr   = kv & 3;
  int lane = n + 16 * (kr >> 1);
  int r    = kr & 1;
  wswz[(size_t)kb * 64 + lane * 2 + r] = acc;
  if (t < NH) {
    float bacc = 0.f;
    for (int k = 0; k < DK; ++k) bacc += k_bias[t * DK + k] * query[t * DK + k];
    blogit[t] = bacc;
  }
}

// ---------------------------------------------------------------------------
// Kernel 2: logits[b][n][s] = x[row]·w[:,n] + blogit[n]
// Skinny GEMM M=32768, K=1024, N=16 via v_wmma_f32_16x16x4_f32.
// One wave per 64 rows (4 M-tiles share each B fragment).
// ---------------------------------------------------------------------------
__global__ void __launch_bounds__(32) logits_kernel(
    const float* __restrict__ x,       // [B*S][KV]
    const float* __restrict__ wswz,    // [KV/4][32][2] swizzled
    const float* __restrict__ blogit,  // [NH]
    float* __restrict__ logits) {      // [B][NH][S]
  const int lane = threadIdx.x;
  const int half = lane >> 4;
  const int lm   = lane & 15;
  const size_t row0 = (size_t)blockIdx.x * 64;

  // A fragment base: lane l, VGPR r -> x[row0 + tile*16 + lm][ks*4 + r + 2*half]
  const float* a0 = x + (row0 + 0 + lm) * KV + half * 2;
  const float* a1 = a0 + (size_t)16 * KV;
  const float* a2 = a0 + (size_t)32 * KV;
  const float* a3 = a0 + (size_t)48 * KV;
  const v2f* wp = reinterpret_cast<const v2f*>(wswz) + lane;

  v8f c0 = {}, c1 = {}, c2 = {}, c3 = {};
  for (int ks = 0; ks < KV / 4; ++ks) {
    v2f bf = wp[ks * 32];
    v2f f0 = *reinterpret_cast<const v2f*>(a0 + ks * 4);
    v2f f1 = *reinterpret_cast<const v2f*>(a1 + ks * 4);
    v2f f2 = *reinterpret_cast<const v2f*>(a2 + ks * 4);
    v2f f3 = *reinterpret_cast<const v2f*>(a3 + ks * 4);
    c0 = WMMA_F32_4(f0, bf, c0);
    c1 = WMMA_F32_4(f1, bf, c1);
    c2 = WMMA_F32_4(f2, bf, c2);
    c3 = WMMA_F32_4(f3, bf, c3);
  }

  const float bl = blogit[lm];
  auto store_tile = [&](v8f c, size_t tile_row0) {
#pragma unroll
    for (int r = 0; r < 8; ++r) {
      size_t row = tile_row0 + r + 8 * half;   // D: M = r + 8*half, N = lm
      size_t b = row >> 10;
      size_t s = row & 1023;
      logits[(b * NH + lm) * S + s] = c[r] + bl;
    }
  };
  store_tile(c0, row0);
  store_tile(c1, row0 + 16);
  store_tile(c2, row0 + 32);
  store_tile(c3, row0 + 48);
}

// ---------------------------------------------------------------------------
// Kernel 3: softmax over s per (b,n); write attn in WMMA-A swizzled layout
//   attnswz[b][(s/4)*64 + lane*2 + r], lane = n + 16*((s%4)/2), r = (s%4)&1
// ---------------------------------------------------------------------------
__global__ void __launch_bounds__(256) softmax_kernel(
    const float* __restrict__ logits,  // [B][NH][S]
    float* __restrict__ attnswz) {     // [B][S/4][32][2]
  const int n = blockIdx.x;
  const int b = blockIdx.y;
  const int t = threadIdx.x;
  const float* L = logits + ((size_t)b * NH + n) * S;
  __shared__ float red[256];

  float vals[4];
  float m = -3.402823466e38f;
#pragma unroll
  for (int i = 0; i < 4; ++i) {
    vals[i] = L[t + 256 * i];
    m = fmaxf(m, vals[i]);
  }
  red[t] = m;
  __syncthreads();
  for (int off = 128; off > 0; off >>= 1) {
    if (t < off) red[t] = fmaxf(red[t], red[t + off]);
    __syncthreads();
  }
  m = red[0];
  __syncthreads();

  float sum = 0.f;
#pragma unroll
  for (int i = 0; i < 4; ++i) {
    vals[i] = __expf(vals[i] - m);
    sum += vals[i];
  }
  red[t] = sum;
  __syncthreads();
  for (int off = 128; off > 0; off >>= 1) {
    if (t < off) red[t] += red[t + off];
    __syncthreads();
  }
  const float inv = 1.f / red[0];

  float* ab = attnswz + (size_t)b * (S / 4) * 64;
#pragma unroll
  for (int i = 0; i < 4; ++i) {
    int s    = t + 256 * i;
    int kr   = s & 3;
    int lane = n + 16 * (kr >> 1);
    int r    = kr & 1;
    ab[(size_t)(s >> 2) * 64 + lane * 2 + r] = vals[i] * inv;
  }
}

// ---------------------------------------------------------------------------
// Kernel 4: y[b][n][kv] = sum_s attn[b][n][s] * x[b][s][kv]
// Per-batch GEMM M=16(heads), K=1024(s), N=1024(kv); one wave per
// (batch, 64 kv columns), 4 N-tiles share each A fragment.
// ---------------------------------------------------------------------------
__global__ void __launch_bounds__(32) weighted_sum_kernel(
    const float* __restrict__ x,        // [B][S][KV]
    const float* __restrict__ attnswz,  // [B][S/4][32][2]
    float* __restrict__ y) {            // [B][NH][KV]
  const int lane = threadIdx.x;
  const int half = lane >> 4;
  const int lm   = lane & 15;
  const int b    = blockIdx.y;
  const int kv0  = blockIdx.x * 64;

  const v2f* ap = reinterpret_cast<const v2f*>(attnswz + (size_t)b * (S / 4) * 64) + lane;
  const float* xb = x + (size_t)b * S * KV + kv0 + lm;

  v8f c0 = {}, c1 = {}, c2 = {}, c3 = {};
  for (int ks = 0; ks < S / 4; ++ks) {
    v2f af = ap[ks * 32];
    // B fragment: lane l, VGPR r -> x[ks*4 + r + 2*half][kv0 + tile*16 + lm]
    const float* xr = xb + (size_t)(ks * 4 + 2 * half) * KV;
    v2f b0 = {xr[0],  xr[KV + 0]};
    v2f b1 = {xr[16], xr[KV + 16]};
    v2f b2 = {xr[32], xr[KV + 32]};
    v2f b3 = {xr[48], xr[KV + 48]};
    c0 = WMMA_F32_4(af, b0, c0);
    c1 = WMMA_F32_4(af, b1, c1);
    c2 = WMMA_F32_4(af, b2, c2);
    c3 = WMMA_F32_4(af, b3, c3);
  }

  float* yb = y + (size_t)b * NH * KV + kv0 + lm;
  auto store_tile = [&](v8f c, int toff) {
#pragma unroll
    for (int r = 0; r < 8; ++r) {
      int n = r + 8 * half;                    // D: M = n, N = kv
      yb[(size_t)n * KV + toff] = c[r];
    }
  };
  store_tile(c0, 0);
  store_tile(c1, 16);
  store_tile(c2, 32);
  store_tile(c3, 48);
}

// ---------------------------------------------------------------------------
// Kernel 5: out[b][n][v] = y[b][n][:]·v_kernel[:, n*64+v] + v_bias[n*64+v]
// 67 MFLOP; v_kernel (4 MB) is L2-resident across batches.
// ---------------------------------------------------------------------------
__global__ void __launch_bounds__(256) out_kernel(
    const float* __restrict__ y,         // [B][NH][KV]
    const float* __restrict__ v_kernel,  // [KV][NH*DV]
    const float* __restrict__ v_bias,    // [NH*DV]
    float* __restrict__ out) {           // [B][NH][DV]
  int t = blockIdx.x * 256 + threadIdx.x;  // [0, B*NH*DV)
  int v = t & 63;
  int n = (t >> 6) & 15;
  int b = t >> 10;
  const float* yp = y + ((size_t)b * NH + n) * KV;
  const float* vp = v_kernel + n * DV + v;
  float acc = 0.f;
#pragma unroll 4
  for (int kv = 0; kv < KV; ++kv) acc += yp[kv] * vp[(size_t)kv * (NH * DV)];
  out[t] = acc + v_bias[n * DV + v];
}

// ---------------------------------------------------------------------------
extern "C" void kernel_launch(void* const* d_in, const int* in_sizes, int n_in,
                              void* d_out, int out_size, void* d_ws, size_t ws_size,
                              hipStream_t stream) {
  const float* key_value = (const float*)d_in[0];  // [32,32,32,1024]
  const float* query     = (const float*)d_in[1];  // [1,1,16,64]
  const float* k_kernel  = (const float*)d_in[2];  // [1024,1024]
  const float* k_bias    = (const float*)d_in[3];  // [1024]
  const float* v_kernel  = (const float*)d_in[4];  // [1024,1024]
  const float* v_bias    = (const float*)d_in[5];  // [1024]
  float* out = (float*)d_out;                      // [32,16,64]

  float* ws      = (float*)d_ws;
  float* wswz    = ws;                  // 16384 floats (64 KB)
  float* blogit  = wswz + 16384;        // 16 floats (padded to 32)
  float* logits  = blogit + 32;         // 524288 floats (2 MB)
  float* attnswz = logits + 524288;     // 524288 floats (2 MB)
  float* ybuf    = attnswz + 524288;    // 524288 floats (2 MB)

  prep_w_kernel<<<64, 256, 0, stream>>>(k_kernel, k_bias, query, wswz, blogit);
  logits_kernel<<<(BATCH * S) / 64, 32, 0, stream>>>(key_value, wswz, blogit, logits);
  softmax_kernel<<<dim3(NH, BATCH), 256, 0, stream>>>(logits, attnswz);
  weighted_sum_kernel<<<dim3(KV / 64, BATCH), 32, 0, stream>>>(key_value, attnswz, ybuf);
  out_kernel<<<(BATCH * NH * DV) / 256, 256, 0, stream>>>(ybuf, v_kernel, v_bias, out);
}